// VGAEEncoder_44813688767186
// MI455X (gfx1250) — compile-verified
//
#include <hip/hip_runtime.h>
#include <hip/hip_bf16.h>

// ---------------------------------------------------------------------------
// VGAE encoder (3x GCNConv) for MI455X / gfx1250.
// GEMMs use V_WMMA_F32_16X16X4_F32 (full fp32 matrix path, matches reference
// precision; GEMM FLOPs are negligible vs the L2-resident scatter/gather).
// Scatter-add uses global_atomic_add_f32: all per-layer arrays (~51MB) fit in
// the 192MB L2, so atomics resolve in-cache.
// ---------------------------------------------------------------------------

typedef float v2f __attribute__((ext_vector_type(2)));
typedef float v8f __attribute__((ext_vector_type(8)));

#define DIN_  128
#define DH_   128
#define DLAT_ 64

// ---------------------------- degree / dinv --------------------------------

__global__ void k_deg_init(float* __restrict__ deg, int n) {
    int i = blockIdx.x * blockDim.x + threadIdx.x;
    if (i < n) deg[i] = 1.0f;               // self-loop contributes 1
}

__global__ void k_deg_edges(const int* __restrict__ dst, float* __restrict__ deg, int e) {
    int i = blockIdx.x * blockDim.x + threadIdx.x;
    if (i < e) atomicAdd(&deg[dst[i]], 1.0f);
}

__global__ void k_rsqrt(float* __restrict__ dinv, int n) {
    int i = blockIdx.x * blockDim.x + threadIdx.x;
    if (i < n) {
        float d = dinv[i];
        dinv[i] = (d > 0.0f) ? rsqrtf(d) : 0.0f;
    }
}

// ------------------------------ WMMA GEMM ----------------------------------
// H[n, o] = sum_k A[n, k] * W[o, k]   (A: nrows x din row-major, W: DOUT x din)
// One wave computes a 16-row M-tile across all DOUT columns.
// f32 WMMA fragment layout (ISA 7.12.2, 32-bit A 16x4):
//   A: lane&15 = row M; lanes>=16 hold K=2,3; per-lane float2 at k + 2*(lane>>4)
//   B: symmetric, lane&15 = column N
//   C/D: VGPR r -> M = r + 8*(lane>=16), N = lane&15
template <int DOUT>
__global__ __launch_bounds__(256)
void k_gemm_xWt(const float* __restrict__ A, const float* __restrict__ W,
                float* __restrict__ H, int nrows, int din) {
    const int lane  = threadIdx.x & 31;
    const int wave  = threadIdx.x >> 5;
    const int mtile = blockIdx.x * 8 + wave;          // wave-uniform
    if (mtile * 16 >= nrows) return;                  // whole wave exits: EXEC stays all-1s

    const int half = lane >> 4;                       // 0 or 1
    const int l16  = lane & 15;
    const int row  = mtile * 16 + l16;
    const int koff = half * 2;

    constexpr int NT = DOUT / 16;
    v8f acc[NT];
#pragma unroll
    for (int t = 0; t < NT; ++t)
#pragma unroll
        for (int r = 0; r < 8; ++r) acc[t][r] = 0.0f;

    const float* arow = A + (size_t)row * din;
    for (int k = 0; k < din; k += 4) {
        v2f a = *(const v2f*)(arow + k + koff);       // 8B-aligned (din%4==0)
#pragma unroll
        for (int t = 0; t < NT; ++t) {
            const int ocol = t * 16 + l16;
            v2f b = *(const v2f*)(W + (size_t)ocol * din + k + koff);
            acc[t] = __builtin_amdgcn_wmma_f32_16x16x4_f32(
                /*neg_a=*/false, a, /*neg_b=*/false, b,
                /*c_mod=*/(short)0, acc[t], /*reuse_a=*/false, /*reuse_b=*/false);
        }
    }

    const int rowbase = mtile * 16 + half * 8;
#pragma unroll
    for (int t = 0; t < NT; ++t) {
#pragma unroll
        for (int r = 0; r < 8; ++r) {
            H[(size_t)(rowbase + r) * DOUT + t * 16 + l16] = acc[t][r];
        }
    }
}

// ------------------------- aggregation kernels -----------------------------

// agg[i,:] = dinv[i]^2 * h[i,:]   (self-loop term; fully initializes agg)
template <int D>
__global__ void k_self_init(const float* __restrict__ dinv, const float* __restrict__ h,
                            float* __restrict__ agg, int n) {
    int t = blockIdx.x * blockDim.x + threadIdx.x;
    if (t >= n * D) return;
    int i = t / D;
    float s = dinv[i];
    agg[t] = s * s * h[t];
}

// agg[dst,:] += dinv[src]*dinv[dst] * h[src,:]
template <int D>
__global__ __launch_bounds__(256)
void k_edge_scatter(const int* __restrict__ src, const int* __restrict__ dst,
                    const float* __restrict__ dinv, const float* __restrict__ h,
                    float* __restrict__ agg, int nedges) {
    constexpr int EPB = 256 / D;                       // edges per block
    int e = blockIdx.x * EPB + threadIdx.x / D;
    if (e >= nedges) return;
    int j = threadIdx.x % D;
    int s = src[e];
    int d = dst[e];
    float nm = dinv[s] * dinv[d];
    atomicAdd(agg + (size_t)d * D + j, nm * h[(size_t)s * D + j]);
}

template <int D, bool RELU>
__global__ void k_bias_act(float* __restrict__ agg, const float* __restrict__ b, int n) {
    int t = blockIdx.x * blockDim.x + threadIdx.x;
    if (t >= n * D) return;
    float v = agg[t] + b[t % D];
    if (RELU) v = fmaxf(v, 0.0f);
    agg[t] = v;
}

// mu += b3 (in place), duplicate into logstd half (reference returns (mu, mu))
__global__ void k_finalize(float* __restrict__ out, const float* __restrict__ b, int n) {
    int t = blockIdx.x * blockDim.x + threadIdx.x;
    if (t >= n * DLAT_) return;
    float v = out[t] + b[t % DLAT_];
    out[t] = v;
    out[(size_t)n * DLAT_ + t] = v;
}

// ------------------------------ launcher -----------------------------------

extern "C" void kernel_launch(void* const* d_in, const int* in_sizes, int n_in,
                              void* d_out, int out_size, void* d_ws, size_t ws_size,
                              hipStream_t stream) {
    const float* x  = (const float*)d_in[0];
    const float* W1 = (const float*)d_in[1];
    const float* b1 = (const float*)d_in[2];
    const float* W2 = (const float*)d_in[3];
    const float* b2 = (const float*)d_in[4];
    const float* W3 = (const float*)d_in[5];
    const float* b3 = (const float*)d_in[6];
    const int*   ei = (const int*)d_in[7];

    const int N = in_sizes[0] / DIN_;
    const int E = in_sizes[7] / 2;
    const int* src = ei;
    const int* dst = ei + E;

    // workspace layout: dinv[N] | bufA[N*128] | bufB[N*128]
    float* dinv = (float*)d_ws;
    size_t off  = (((size_t)N * sizeof(float)) + 255) & ~(size_t)255;
    float* bufA = (float*)((char*)d_ws + off);
    float* bufB = bufA + (size_t)N * DH_;
    float* mu   = (float*)d_out;

    const int T = 256;
    const int mblocks = (N / 16 + 7) / 8;              // 8 waves (M-tiles) per block
    const int gridE128 = (E + 1) / 2;                  // 2 edges per 256-thr block
    const int gridE64  = (E + 3) / 4;                  // 4 edges per block

    // degree -> dinv
    k_deg_init <<<(N + T - 1) / T, T, 0, stream>>>(dinv, N);
    k_deg_edges<<<(E + T - 1) / T, T, 0, stream>>>(dst, dinv, E);
    k_rsqrt    <<<(N + T - 1) / T, T, 0, stream>>>(dinv, N);

    // ---- layer 1: h = relu(scatter(norm * (x @ W1^T)) + b1)
    k_gemm_xWt<DH_><<<mblocks, T, 0, stream>>>(x, W1, bufA, N, DIN_);
    k_self_init<DH_><<<((size_t)N * DH_ + T - 1) / T, T, 0, stream>>>(dinv, bufA, bufB, N);
    k_edge_scatter<DH_><<<gridE128, T, 0, stream>>>(src, dst, dinv, bufA, bufB, E);
    k_bias_act<DH_, true><<<((size_t)N * DH_ + T - 1) / T, T, 0, stream>>>(bufB, b1, N);

    // ---- layer 2
    k_gemm_xWt<DH_><<<mblocks, T, 0, stream>>>(bufB, W2, bufA, N, DH_);
    k_self_init<DH_><<<((size_t)N * DH_ + T - 1) / T, T, 0, stream>>>(dinv, bufA, bufB, N);
    k_edge_scatter<DH_><<<gridE128, T, 0, stream>>>(src, dst, dinv, bufA, bufB, E);
    k_bias_act<DH_, true><<<((size_t)N * DH_ + T - 1) / T, T, 0, stream>>>(bufB, b2, N);

    // ---- layer 3 (mu): aggregate straight into d_out[0 : N*64)
    k_gemm_xWt<DLAT_><<<mblocks, T, 0, stream>>>(bufB, W3, bufA, N, DH_);
    k_self_init<DLAT_><<<((size_t)N * DLAT_ + T - 1) / T, T, 0, stream>>>(dinv, bufA, mu, N);
    k_edge_scatter<DLAT_><<<gridE64, T, 0, stream>>>(src, dst, dinv, bufA, mu, E);
    k_finalize<<<((size_t)N * DLAT_ + T - 1) / T, T, 0, stream>>>(mu, b3, N);
}